// DiscreteContinuousEncoder_6614249636089
// MI455X (gfx1250) — compile-verified
//
#include <hip/hip_runtime.h>

typedef __attribute__((ext_vector_type(2))) float v2f;
typedef __attribute__((ext_vector_type(8))) float v8f;

#define B_    4
#define CIN   8
#define COUT  32
#define HIN   721
#define WIN   1440
#define HOUT  361
#define WOUT  720
#define K_    9
#define NNZ   32
#define TPO   48                    // po tile (multiple of 16, divides 720)
#define NTILES (WOUT / TPO)         // 15
#define SLABW (2 * TPO + 16)        // 112 x-columns needed per tile
#define NROWS 5                     // rows 2h-2 .. 2h+2 (clamped)
#define CK    (CIN * K_)            // 72 = GEMM K dimension
#define JCH   (TPO / 4)             // 12 output columns per phase-1 task
#define NTHREADS (CK * 4)           // 288 threads = 9 waves; 1 task/thread in phase 1

__global__ __launch_bounds__(NTHREADS) void disco_fused(
    const float* __restrict__ x, const int* __restrict__ psi_idx,
    const float* __restrict__ psi_vals, const float* __restrict__ w,
    float* __restrict__ out)
{
  __shared__ float xs[NROWS * CIN * SLABW];   // 17,920 B  x slab
  __shared__ v2f   ypk[(CK / 2) * TPO];       // 13,824 B  y packed: {y[2k2][j], y[2k2+1][j]}
  __shared__ v2f   wpk[COUT * (CK / 2)];      //  9,216 B  W packed: {W[o][2k2], W[o][2k2+1]}
  __shared__ int   offs[K_ * NNZ];            //  1,152 B  packed gather offsets
  __shared__ float vals[K_ * NNZ];            //  1,152 B  psi values

  const int tid  = threadIdx.x;
  int bid        = blockIdx.x;
  const int tile = bid % NTILES;  bid /= NTILES;
  const int h    = bid % HOUT;
  const int b    = bid / HOUT;
  const int po0  = tile * TPO;

  // ---- Phase 0: psi metadata + weight + x slab into LDS ----
  for (int i = tid; i < K_ * NNZ; i += NTHREADS) {
    int k = i >> 5, n = i & 31;
    int gi  = (k * HOUT + h) * NNZ + n;
    int id  = psi_idx[gi];
    int row = id / WIN;
    int col = id - row * WIN;          // in [0,16)
    int rl  = row - 2 * h + 2;         // local row in [0,4] (base_row[h] == 2h exactly)
    offs[i] = rl * (CIN * SLABW) + col;
    vals[i] = psi_vals[gi];
  }
  {
    float* wf = (float*)wpk;           // (COUT,CIN,K_) is o-major; ck = c*9+k matches y
    for (int i = tid; i < COUT * CK; i += NTHREADS) wf[i] = w[i];
  }
  for (int i = tid; i < NROWS * CIN * SLABW; i += NTHREADS) {
    int tcol = i % SLABW;
    int rc   = i / SLABW;              // ri*CIN + c
    int c    = rc % CIN;
    int ri   = rc / CIN;
    int g    = (2 * po0 + tcol) % WIN; // column wrap handled here only
    int grow = 2 * h - 2 + ri;
    grow = grow < 0 ? 0 : (grow > HIN - 1 ? HIN - 1 : grow);
    xs[i] = x[(((size_t)b * CIN + c) * HIN + grow) * WIN + g];
  }
  __syncthreads();

  // ---- Phase 1: one task per thread: y[ck][jc*12 .. jc*12+11] ----
  {
    const int ck = tid >> 2;           // 0..71
    const int jc = tid & 3;            // 0..3
    const int c  = ck / K_;
    const int k  = ck - c * K_;
    const int base = c * SLABW + 2 * (jc * JCH);
    const int*   op = &offs[k * NNZ];
    const float* vp = &vals[k * NNZ];
    float acc[JCH];
#pragma unroll
    for (int jj = 0; jj < JCH; ++jj) acc[jj] = 0.f;
    for (int n = 0; n < NNZ; ++n) {    // 2 uniform LDS loads + 12 gathers per iter
      const int   o = op[n];
      const float v = vp[n];
#pragma unroll
      for (int jj = 0; jj < JCH; ++jj)
        acc[jj] += v * xs[o + base + 2 * jj];
    }
    float* yf = (float*)ypk;           // packed: index = k2*(2*TPO) + 2*j + (ck&1)
#pragma unroll
    for (int jj = 0; jj < JCH; ++jj)
      yf[(ck >> 1) * (2 * TPO) + 2 * (jc * JCH + jj) + (ck & 1)] = acc[jj];
  }
  __syncthreads();

  // ---- Phase 2: out(32 x 48) = W(32 x 72) * Y(72 x 48) via V_WMMA_F32_16X16X4_F32 ----
  const int wave = tid >> 5;
  const int lane = tid & 31;
  if (wave < 6) {                      // 6 waves, one 16x16 D tile each (2 M x 3 N)
    const int mt  = wave / 3;
    const int nt  = wave - mt * 3;
    const int n16 = lane & 15;
    const int hl  = lane >> 4;         // lanes 16-31 carry K+2/K+3 and M+8
    v8f acc = {};
#pragma unroll
    for (int kk = 0; kk < CK / 4; ++kk) {
      const int k2 = kk * 2 + hl;      // float2 pair index: {K=4kk+2hl, K=4kk+2hl+1}
      v2f a  = wpk[(mt * 16 + n16) * (CK / 2) + k2];  // single ds_load_b64
      v2f bb = ypk[k2 * TPO + nt * 16 + n16];         // single ds_load_b64, no bank conflicts
      acc = __builtin_amdgcn_wmma_f32_16x16x4_f32(
          /*neg_a=*/false, a, /*neg_b=*/false, bb,
          /*c_mod=*/(short)0, acc, /*reuse_a=*/false, /*reuse_b=*/false);
    }
    // D layout: VGPR v -> M = v + 8*hl, N = lane&15
    const int po = po0 + nt * 16 + n16;
#pragma unroll
    for (int v = 0; v < 8; ++v) {
      int o = mt * 16 + v + 8 * hl;
      out[(((size_t)b * COUT + o) * HOUT + h) * WOUT + po] = acc[v];
    }
  }
}

extern "C" void kernel_launch(void* const* d_in, const int* in_sizes, int n_in,
                              void* d_out, int out_size, void* d_ws, size_t ws_size,
                              hipStream_t stream) {
  const float* x        = (const float*)d_in[0];
  const int*   psi_idx  = (const int*)  d_in[1];
  const float* psi_vals = (const float*)d_in[2];
  const float* w        = (const float*)d_in[3];
  float*       out      = (float*)d_out;

  dim3 grid(B_ * HOUT * NTILES);       // 4 * 361 * 15 = 21660 blocks
  disco_fused<<<grid, NTHREADS, 0, stream>>>(x, psi_idx, psi_vals, w, out);
}